// Encoder_7524782702660
// MI455X (gfx1250) — compile-verified
//
#include <hip/hip_runtime.h>

// LSTM encoder: V=32000, D=H=1024, L=2, B=64, S=256
// Strategy:
//   per layer:  xg = x @ Wx + b      (big WMMA GEMM, gather fused for layer 0)
//   per step :  gates = h @ Wh + xg[t]  (small WMMA GEMM)
//               pointwise sigmoid/tanh + masked (c,h) update
// fp32 matrix path: V_WMMA_F32_16X16X4_F32 (keeps reference precision).
// Data path: GLOBAL_LOAD_ASYNC_TO_LDS_B128 double-buffered (ASYNCcnt),
//            one barrier per K-slice.

typedef __attribute__((ext_vector_type(2))) float v2f;
typedef __attribute__((ext_vector_type(8))) float v8f;

#define LSTM_B 64
#define LSTM_S 256
#define LSTM_D 1024
#define LSTM_H 1024
#define LSTM_L 2
#define GN     4096   // 4*H

#define TM 64
#define TN 64
#define BK 16
#define AST 18        // padded LDS row stride for A tile (16 + 2)

// 32-bit LDS byte offset for the async-copy VDST operand (generic-pointer
// low 32 bits == wave-relative LDS offset per the LDS aperture mapping).
__device__ __forceinline__ unsigned lds_off32(const void* p)
{
    return (unsigned)(uintptr_t)p;
}

// Async global -> LDS copy, 16 bytes per lane, tracked by ASYNCcnt.
__device__ __forceinline__ void async_copy_b128(unsigned lds_byte_off,
                                                const float* gptr)
{
    asm volatile("global_load_async_to_lds_b128 %0, %1, off"
                 :: "v"(lds_byte_off), "v"(gptr)
                 : "memory");
}

__device__ __forceinline__ void wait_async0()
{
    asm volatile("s_wait_asynccnt 0x0" ::: "memory");
}

// mode: 0 = direct rows   (rowptr = A + m*K)            [h @ Wh]
//       1 = seq rows      (rowptr = A + (b*S+s)*K)      [x1 @ Wx1], m = s*B+b
//       2 = token gather  (rowptr = A + tokens[b*S+s]*K)[emb @ Wx0]
__global__ __launch_bounds__(256)
void lstm_gemm_wmma_f32(const float* __restrict__ A,
                        const int*   __restrict__ tokens,
                        const float* __restrict__ Bw,     // [K][N]
                        const float* __restrict__ bias,   // [N] or null
                        const float* __restrict__ addend, // [M][N] slab or null
                        float* __restrict__ out,          // [M][N]
                        int M, int N, int K, int mode)
{
    __shared__ float As[2][TM * AST];
    __shared__ float Bs[2][BK * TN];

    const int tid  = threadIdx.x;
    const int m0   = blockIdx.x * TM;
    const int n0   = blockIdx.y * TN;

    // ---- A-tile mapping: 64 rows x 16 cols, one B128 per thread ----
    const int lr     = tid >> 2;          // 0..63 local row
    const int achunk = (tid & 3) * 4;     // 0,4,8,12
    const int gm     = m0 + lr;
    const float* rowA;
    if (mode == 2) {
        int s = gm / LSTM_B, b = gm % LSTM_B;
        rowA = A + (size_t)tokens[b * LSTM_S + s] * (size_t)K;
    } else if (mode == 1) {
        int s = gm / LSTM_B, b = gm % LSTM_B;
        rowA = A + ((size_t)b * LSTM_S + s) * (size_t)K;
    } else {
        rowA = A + (size_t)gm * (size_t)K;
    }

    // ---- B-tile mapping: 16 rows x 64 cols, one B128 per thread ----
    const int bk     = tid >> 4;          // 0..15 local k row
    const int bchunk = (tid & 15) * 4;    // 0..60
    const float* rowB = Bw + (size_t)bk * (size_t)N + n0 + bchunk;

    const unsigned ldsA[2] = { lds_off32(&As[0][lr * AST + achunk]),
                               lds_off32(&As[1][lr * AST + achunk]) };
    const unsigned ldsB[2] = { lds_off32(&Bs[0][bk * TN + bchunk]),
                               lds_off32(&Bs[1][bk * TN + bchunk]) };

    // ---- wave -> output subtile mapping ----
    const int lane = tid & 31;
    const int w    = tid >> 5;            // 0..7
    const int half = lane >> 4;           // 0/1
    const int r    = lane & 15;           // 0..15
    const int rm   = (w & 3) * 16;        // subtile row base within TM
    const int cn   = (w >> 2) * 32;       // subtile col base within TN

    v8f acc0 = {};
    v8f acc1 = {};

    const int nk = K / BK;

    // prologue: stage 0 into buffer 0
    async_copy_b128(ldsA[0], rowA + achunk);
    async_copy_b128(ldsB[0], rowB);

    for (int kt = 0; kt < nk; ++kt) {
        const int cur = kt & 1;

        wait_async0();        // my stage-kt writes to LDS are done
        __syncthreads();      // everyone's done writing kt / computing kt-1

        if (kt + 1 < nk) {    // prefetch stage kt+1 into the other buffer
            const int k0n = (kt + 1) * BK;
            async_copy_b128(ldsA[cur ^ 1], rowA + k0n + achunk);
            async_copy_b128(ldsB[cur ^ 1],
                            rowB + (size_t)k0n * (size_t)N);
        }

        const float* __restrict__ Ac = &As[cur][0];
        const float* __restrict__ Bc = &Bs[cur][0];
#pragma unroll
        for (int ks = 0; ks < 4; ++ks) {
            const int kk = ks * 4 + 2 * half;  // per ISA A 16x4 / B 4x16 layouts
            v2f a;
            a.x = Ac[(rm + r) * AST + kk];
            a.y = Ac[(rm + r) * AST + kk + 1];
            v2f b0;
            b0.x = Bc[(kk    ) * TN + cn + r];
            b0.y = Bc[(kk + 1) * TN + cn + r];
            v2f b1;
            b1.x = Bc[(kk    ) * TN + cn + 16 + r];
            b1.y = Bc[(kk + 1) * TN + cn + 16 + r];
            acc0 = __builtin_amdgcn_wmma_f32_16x16x4_f32(
                false, a, false, b0, (short)0, acc0, false, false);
            acc1 = __builtin_amdgcn_wmma_f32_16x16x4_f32(
                false, a, false, b1, (short)0, acc1, false, false);
        }
        // no trailing barrier: next iteration's wait+barrier protects reuse
    }

    // ---- epilogue: D element (vg + 8*half, r) per ISA C/D 16x16 layout ----
#pragma unroll
    for (int vg = 0; vg < 8; ++vg) {
        const int row = m0 + rm + vg + 8 * half;
        const int c0  = n0 + cn + r;
        float v0 = acc0[vg];
        float v1 = acc1[vg];
        if (bias) { v0 += bias[c0]; v1 += bias[c0 + 16]; }
        if (addend) {
            v0 += addend[(size_t)row * N + c0];
            v1 += addend[(size_t)row * N + c0 + 16];
        }
        out[(size_t)row * N + c0]      = v0;
        out[(size_t)row * N + c0 + 16] = v1;
    }
}

__global__ __launch_bounds__(256)
void lstm_pointwise(const float* __restrict__ gates,   // [B][4H]
                    const int*   __restrict__ lengths, // [B]
                    int t,
                    float* __restrict__ c_state,       // [B][H]
                    float* __restrict__ h_state,       // [B][H]
                    float* __restrict__ hseq,          // [B][S][H]
                    int S)
{
    const int idx = blockIdx.x * blockDim.x + threadIdx.x;  // B*H
    const int b = idx / LSTM_H;
    const int n = idx % LSTM_H;
    const float* g = gates + (size_t)b * GN;
    const float gi = g[n];
    const float gf = g[n + LSTM_H];
    const float gg = g[n + 2 * LSTM_H];
    const float go = g[n + 3 * LSTM_H];

    const float si = 1.0f / (1.0f + expf(-gi));
    const float sf = 1.0f / (1.0f + expf(-gf));
    const float so = 1.0f / (1.0f + expf(-go));

    const float c_old = c_state[idx];
    const float c_new = sf * c_old + si * tanhf(gg);
    const float h_new = so * tanhf(c_new);

    const bool live = t < lengths[b];
    const float c = live ? c_new : c_old;
    const float h = live ? h_new : h_state[idx];
    c_state[idx] = c;
    h_state[idx] = h;
    hseq[((size_t)b * S + t) * LSTM_H + n] = h;
}

__global__ void lstm_zero(float* __restrict__ p, int n)
{
    int i = blockIdx.x * blockDim.x + threadIdx.x;
    if (i < n) p[i] = 0.0f;
}

extern "C" void kernel_launch(void* const* d_in, const int* in_sizes, int n_in,
                              void* d_out, int out_size, void* d_ws, size_t ws_size,
                              hipStream_t stream)
{
    (void)in_sizes; (void)n_in; (void)out_size; (void)ws_size;
    const int*   inputs  = (const int*)d_in[0];   // [B,S]
    const int*   lengths = (const int*)d_in[1];   // [B]
    /* d_in[2] = train flag, unused (dropout rates are 0) */
    const float* emb     = (const float*)d_in[3]; // [V,D]
    const float* Wx      = (const float*)d_in[4]; // [L,D,4H]
    const float* Wh      = (const float*)d_in[5]; // [L,H,4H]
    const float* bias    = (const float*)d_in[6]; // [L,4H]

    float* out = (float*)d_out;                   // x [B,S,H] | cs [L,B,H] | hs [L,B,H]
    const size_t BSH = (size_t)LSTM_B * LSTM_S * LSTM_H;
    const size_t BH  = (size_t)LSTM_B * LSTM_H;
    float* cs = out + BSH;
    float* hs = cs + (size_t)LSTM_L * BH;

    // workspace: xg [S,B,4H] (256MB) | x1 [B,S,H] (64MB) | gates [B,4H] (1MB)
    float* xg    = (float*)d_ws;
    float* x1    = xg + (size_t)LSTM_S * LSTM_B * GN;
    float* gates = x1 + BSH;

    // zero the (c,h) state regions (they live directly in d_out's tail)
    {
        const int n = 2 * LSTM_L * (int)BH;
        lstm_zero<<<(n + 255) / 256, 256, 0, stream>>>(cs, n);
    }

    const int M_xg = LSTM_B * LSTM_S;  // 16384
    for (int l = 0; l < LSTM_L; ++l) {
        const float* Ain   = (l == 0) ? emb : x1;
        const int*   tok   = (l == 0) ? inputs : nullptr;
        const int    mode  = (l == 0) ? 2 : 1;
        const float* Wxl   = Wx + (size_t)l * LSTM_D * GN;
        const float* Whl   = Wh + (size_t)l * LSTM_H * GN;
        const float* bl    = bias + (size_t)l * GN;
        float* c_st = cs + (size_t)l * BH;
        float* h_st = hs + (size_t)l * BH;
        float* hseq = (l == 0) ? x1 : out;

        // xg[s*B+b][g] = x @ Wx + b   (gather fused for layer 0)
        lstm_gemm_wmma_f32<<<dim3(M_xg / TM, GN / TN), 256, 0, stream>>>(
            Ain, tok, Wxl, bl, nullptr, xg, M_xg, GN, LSTM_D, mode);

        for (int t = 0; t < LSTM_S; ++t) {
            // gates = h @ Wh + xg[t]
            lstm_gemm_wmma_f32<<<dim3(LSTM_B / TM, GN / TN), 256, 0, stream>>>(
                h_st, nullptr, Whl, nullptr, xg + (size_t)t * LSTM_B * GN,
                gates, LSTM_B, GN, LSTM_H, 0);
            lstm_pointwise<<<(LSTM_B * LSTM_H) / 256, 256, 0, stream>>>(
                gates, lengths, t, c_st, h_st, hseq, LSTM_S);
        }
    }
}